// MetricEvolutionModel_17480516894903
// MI455X (gfx1250) — compile-verified
//
#include <hip/hip_runtime.h>
#include <math.h>

// Problem constants from the reference
#define B_SZ   256
#define L_SZ   2048
#define EPS_FD 1e-4f
#define DIAGMIN 0.1f
#define NP_SZ  36

typedef float v2f __attribute__((ext_vector_type(2)));
typedef float v8f __attribute__((ext_vector_type(8)));

__constant__ int c_tr[36] = {0,1,1,2,2,2,3,3,3,3,4,4,4,4,4,5,5,5,5,5,5,6,6,6,6,6,6,6,7,7,7,7,7,7,7,7};
__constant__ int c_tc[36] = {0,0,1,0,1,2,0,1,2,3,0,1,2,3,4,0,1,2,3,4,5,0,1,2,3,4,5,6,0,1,2,3,4,5,6,7};

__device__ __forceinline__ float siluf(float x) { return x / (1.f + expf(-x)); }

// Lower-triangular element of the Cholesky-like factor from metric-net output row.
// diag -> softplus(p)+0.1, offdiag -> p
__device__ __forceinline__ float lval(const float* Prow, int r, int c) {
  float p = Prow[(r * (r + 1)) / 2 + c];
  if (r == c) {
    float sp = (p > 20.f) ? p : log1pf(expf(p));
    p = sp + DIAGMIN;
  }
  return p;
}

__global__ __launch_bounds__(32) void metric_evo_kernel(
    const float* __restrict__ values,
    const float* __restrict__ se_in_w,  const float* __restrict__ se_in_b,
    const float* __restrict__ se_conv_w,const float* __restrict__ se_conv_b,
    const float* __restrict__ se_out_w, const float* __restrict__ se_out_b,
    const float* __restrict__ me_w1,    const float* __restrict__ me_b1,
    const float* __restrict__ me_w2,    const float* __restrict__ me_b2,
    const float* __restrict__ ef_w1,    const float* __restrict__ ef_b1,
    const float* __restrict__ ef_w2,    const float* __restrict__ ef_b2,
    const float* __restrict__ ef_scale,
    const float* __restrict__ mu_w1,    const float* __restrict__ mu_b1,
    const float* __restrict__ mu_w2,    const float* __restrict__ mu_b2,
    float* __restrict__ out)
{
  // ---- LDS ----
  __shared__ float sw_in[16], sb_in[16];
  __shared__ float sw_conv[8 * 16 * 3], sb_conv[8];
  __shared__ float sw_out[64], sb_out[8];
  __shared__ float sme_w1[16 * 8], sme_b1[16];
  __shared__ float sme_w2p[48 * 16], sme_b2p[48];   // N padded 36 -> 48
  __shared__ float sef_w1[16 * 16], sef_b1[16];
  __shared__ float sef_w2[8 * 16], sef_b2[8];
  __shared__ float smu_w1[64], smu_b1[8], smu_w2[8], smu_b2s[1];
  __shared__ float sY1[32 * 16];                    // layer-1 activations (silu)
  __shared__ float sP[32 * 48];                     // layer-2 outputs (metric params)
  __shared__ float sG[17 * 64];                     // g matrices for 17 evals
  __shared__ float saug[8 * 16];                    // Gauss-Jordan workspace
  __shared__ float sginv[64];
  __shared__ float szs[16], szmat[16];
  __shared__ float ssl[8], stsh[16], sd2[8], shm[8];

  const int lane = threadIdx.x;
  const int b    = blockIdx.x;
  const int hi16 = lane >> 4;     // 0: lanes 0-15, 1: lanes 16-31
  const int l15  = lane & 15;

  // ---- stage weights to LDS ----
  for (int i = lane; i < 16; i += 32) { sw_in[i] = se_in_w[i]; sb_in[i] = se_in_b[i];
                                        sme_b1[i] = me_b1[i];  sef_b1[i] = ef_b1[i]; }
  for (int i = lane; i < 8 * 16 * 3; i += 32) sw_conv[i] = se_conv_w[i];
  for (int i = lane; i < 8; i += 32)  { sb_conv[i] = se_conv_b[i]; sb_out[i] = se_out_b[i];
                                        sef_b2[i] = ef_b2[i]; smu_b1[i] = mu_b1[i]; smu_w2[i] = mu_w2[i]; }
  for (int i = lane; i < 64; i += 32) { sw_out[i] = se_out_w[i]; smu_w1[i] = mu_w1[i]; }
  for (int i = lane; i < 128; i += 32){ sme_w1[i] = me_w1[i]; sef_w2[i] = ef_w2[i]; }
  for (int i = lane; i < 256; i += 32) sef_w1[i] = ef_w1[i];
  for (int i = lane; i < 48 * 16; i += 32) {
    int n = i >> 4, k = i & 15;
    sme_w2p[i] = (n < NP_SZ) ? me_w2[n * 16 + k] : 0.f;
  }
  for (int i = lane; i < 48; i += 32) sme_b2p[i] = (i < NP_SZ) ? me_b2[i] : 0.f;
  if (lane == 0) smu_b2s[0] = mu_b2[0];

  // only the last 4 sequence values matter (causal conv, k=3, outputs at L-2, L-1)
  const float* vp = values + (size_t)b * L_SZ + (L_SZ - 4);
  const float v0 = vp[0], v1 = vp[1], v2 = vp[2], v3 = vp[3];
  __syncthreads();

  // ---- front-end: h = in-proj, causal conv (taps t-2..t), silu ----
  if (lane < 16) {
    const int o = lane & 7;
    const int tau = 2 + (lane >> 3);               // local timestep 2 (L-2) or 3 (L-1)
    const float vv[4] = {v0, v1, v2, v3};
    float acc = sb_conv[o];
    for (int i = 0; i < 16; ++i) {
      const float wi = sw_in[i], bi = sb_in[i];
      #pragma unroll
      for (int k = 0; k < 3; ++k) {
        const float hv = vv[tau - 2 + k] * wi + bi;
        acc += sw_conv[o * 48 + i * 3 + k] * hv;
      }
    }
    szs[(tau - 2) * 8 + o] = siluf(acc);
  }
  __syncthreads();
  if (lane < 16) {                                 // z = silu(h2) @ se_out_w^T + b
    const int j = lane & 7, ti = lane >> 3;
    float zv = sb_out[j];
    #pragma unroll
    for (int o = 0; o < 8; ++o) zv += sw_out[j * 8 + o] * szs[ti * 8 + o];
    szmat[ti * 8 + j] = zv;                        // row 0 = z_prev, row 1 = z_last
  }
  __syncthreads();

  float z[8], zp[8], dz[8];
  #pragma unroll
  for (int j = 0; j < 8; ++j) { z[j] = szmat[8 + j]; zp[j] = szmat[j]; dz[j] = z[j] - zp[j]; }

  // ---- metric MLP, 32x8 input tile via V_WMMA_F32_16X16X4_F32 ----
  // rows 0-7: z+eps*e_k, rows 8-15: z-eps*e_k, row 16: z, rows 17-31: don't-care
  // A layout (ISA 16x4 f32): lane<16 holds K={0,1}, lane>=16 holds K={2,3} of its row
  auto xval = [&](int row, int c) -> float {
    float v = szmat[8 + c];
    if (row < 8)       { if (c == row)     v += EPS_FD; }
    else if (row < 16) { if (c == row - 8) v -= EPS_FD; }
    return v;
  };

  // layer 1: (32x8) @ me_w1^T(8x16) + b1, then silu
  #pragma unroll
  for (int m = 0; m < 2; ++m) {
    v8f c;
    #pragma unroll
    for (int r = 0; r < 8; ++r) c[r] = sme_b1[l15];
    #pragma unroll
    for (int kc = 0; kc < 2; ++kc) {
      const int k0 = kc * 4 + hi16 * 2;
      const int row = m * 16 + l15;
      v2f a, bv;
      a.x  = xval(row, k0);           a.y  = xval(row, k0 + 1);
      bv.x = sme_w1[l15 * 8 + k0];    bv.y = sme_w1[l15 * 8 + k0 + 1];
      c = __builtin_amdgcn_wmma_f32_16x16x4_f32(false, a, false, bv, (short)0, c, false, false);
    }
    #pragma unroll
    for (int r = 0; r < 8; ++r)       // silu + spill in C/D layout (row = r + 8*hi16, col = l15)
      sY1[(m * 16 + r + 8 * hi16) * 16 + l15] = siluf(c[r]);
  }
  __syncthreads();

  // layer 2: (32x16) @ me_w2^T(16x48 padded) + b2
  #pragma unroll
  for (int m = 0; m < 2; ++m) {
    #pragma unroll
    for (int nt = 0; nt < 3; ++nt) {
      v8f c;
      #pragma unroll
      for (int r = 0; r < 8; ++r) c[r] = sme_b2p[nt * 16 + l15];
      #pragma unroll
      for (int kc = 0; kc < 4; ++kc) {
        const int k0 = kc * 4 + hi16 * 2;
        const int row = m * 16 + l15;
        v2f a, bv;
        a.x  = sY1[row * 16 + k0];                      a.y  = sY1[row * 16 + k0 + 1];
        bv.x = sme_w2p[(nt * 16 + l15) * 16 + k0];      bv.y = sme_w2p[(nt * 16 + l15) * 16 + k0 + 1];
        c = __builtin_amdgcn_wmma_f32_16x16x4_f32(false, a, false, bv, (short)0, c, false, false);
      }
      #pragma unroll
      for (int r = 0; r < 8; ++r)
        sP[(m * 16 + r + 8 * hi16) * 48 + nt * 16 + l15] = c[r];
    }
  }
  __syncthreads();

  // ---- g = L L^T for the 17 used evals (lower tri tasks, mirrored) ----
  for (int tt = lane; tt < 17 * 36; tt += 32) {
    const int e = tt / 36, t = tt % 36;
    const int i = c_tr[t], j = c_tc[t];               // i >= j
    const float* Prow = &sP[e * 48];
    float gsum = 0.f;
    for (int m = 0; m <= j; ++m) gsum += lval(Prow, i, m) * lval(Prow, j, m);
    sG[e * 64 + i * 8 + j] = gsum;
    sG[e * 64 + j * 8 + i] = gsum;
  }
  __syncthreads();

  // ---- invert g(z_last) (SPD -> pivot-free Gauss-Jordan, lanes 0..7 = rows) ----
  if (lane < 8) {
    for (int cq = 0; cq < 8; ++cq) saug[lane * 16 + cq] = sG[16 * 64 + lane * 8 + cq];
    for (int cq = 0; cq < 8; ++cq) saug[lane * 16 + 8 + cq] = (cq == lane) ? 1.f : 0.f;
  }
  __syncthreads();
  for (int p = 0; p < 8; ++p) {
    if (lane < 8 && lane != p) {
      const float f = saug[lane * 16 + p] / saug[p * 16 + p];
      for (int cq = 0; cq < 16; ++cq) saug[lane * 16 + cq] -= f * saug[p * 16 + cq];
    }
    __syncthreads();
  }
  if (lane < 8) {
    const float dinv = 1.f / saug[lane * 16 + lane];
    for (int cq = 0; cq < 8; ++cq) sginv[lane * 8 + cq] = saug[lane * 16 + 8 + cq] * dinv;
  }
  __syncthreads();

  // ---- Christoffel contraction: s_l = (2u_l - w_l)/(2eps), a_geo = -0.5 ginv s ----
  if (lane < 8) {
    const int l = lane;
    float u = 0.f, w = 0.f;
    #pragma unroll
    for (int i = 0; i < 8; ++i) {
      float ui = 0.f;
      const float* gp = &sG[i * 64];
      const float* gm = &sG[(8 + i) * 64];
      #pragma unroll
      for (int j = 0; j < 8; ++j) ui += dz[j] * (gp[j * 8 + l] - gm[j * 8 + l]);
      u += dz[i] * ui;
    }
    {
      const float* gp = &sG[l * 64];
      const float* gm = &sG[(8 + l) * 64];
      #pragma unroll
      for (int i = 0; i < 8; ++i) {
        float wi = 0.f;
        #pragma unroll
        for (int j = 0; j < 8; ++j) wi += dz[j] * (gp[i * 8 + j] - gm[i * 8 + j]);
        w += dz[i] * wi;
      }
    }
    ssl[l] = (2.f * u - w) * (1.f / (2.f * EPS_FD));
  }
  __syncthreads();

  // ---- external force MLP (tanh) ----
  if (lane < 16) {
    float t = sef_b1[lane];
    #pragma unroll
    for (int m = 0; m < 16; ++m) {
      const float fin = (m < 8) ? z[m] : dz[m - 8];
      t += sef_w1[lane * 16 + m] * fin;
    }
    stsh[lane] = tanhf(t);
  }
  __syncthreads();
  if (lane < 8) {
    float F = sef_b2[lane];
    #pragma unroll
    for (int n = 0; n < 16; ++n) F += sef_w2[lane * 16 + n] * stsh[n];
    F *= ef_scale[0];
    float ag = 0.f;
    #pragma unroll
    for (int l2 = 0; l2 < 8; ++l2) ag += sginv[lane * 8 + l2] * ssl[l2];
    sd2[lane] = -0.5f * ag + F;                       // d2z = a_geo + F_ext
  }
  __syncthreads();

  // ---- mu head ----
  if (lane < 8) {
    float hmu = smu_b1[lane];
    #pragma unroll
    for (int m = 0; m < 8; ++m) hmu += smu_w1[lane * 8 + m] * sd2[m];
    shm[lane] = siluf(hmu);
  }
  __syncthreads();
  if (lane == 0) {
    float mu = smu_b2s[0];
    #pragma unroll
    for (int n = 0; n < 8; ++n) mu += smu_w2[n] * shm[n];
    out[b] = v3 + (v3 - v2) + mu;                     // x_last + dx_last + mu
  }
}

extern "C" void kernel_launch(void* const* d_in, const int* in_sizes, int n_in,
                              void* d_out, int out_size, void* d_ws, size_t ws_size,
                              hipStream_t stream) {
  (void)in_sizes; (void)n_in; (void)d_ws; (void)ws_size; (void)out_size;
  metric_evo_kernel<<<B_SZ, 32, 0, stream>>>(
      (const float*)d_in[0],
      (const float*)d_in[1],  (const float*)d_in[2],
      (const float*)d_in[3],  (const float*)d_in[4],
      (const float*)d_in[5],  (const float*)d_in[6],
      (const float*)d_in[7],  (const float*)d_in[8],
      (const float*)d_in[9],  (const float*)d_in[10],
      (const float*)d_in[11], (const float*)d_in[12],
      (const float*)d_in[13], (const float*)d_in[14],
      (const float*)d_in[15],
      (const float*)d_in[16], (const float*)d_in[17],
      (const float*)d_in[18], (const float*)d_in[19],
      (float*)d_out);
}